// ConsistencyConstraint_9543417331727
// MI455X (gfx1250) — compile-verified
//
#include <hip/hip_runtime.h>
#include <math.h>

// Problem constants (from reference)
#define BN 4096      // batch
#define DN 4096      // C*H*W = 1*64*64
#define NCLS 10

typedef __attribute__((ext_vector_type(2))) float v2f;
typedef __attribute__((ext_vector_type(8))) float v8f;

// ---------------------------------------------------------------------------
// K1: per-row min/max/sum/sumsq over grad -> alpha[i], beta[i]
//     vn_i[d] = alpha_i * g[i,d] - beta_i
// ---------------------------------------------------------------------------
__global__ void k1_row_stats(const float* __restrict__ g,
                             float* __restrict__ alpha,
                             float* __restrict__ beta) {
    const int row = blockIdx.x;
    const int t   = threadIdx.x;              // 256 threads
    const float* r = g + (size_t)row * DN;

    float mn = INFINITY, mx = -INFINITY, sm = 0.f, sq = 0.f;
    for (int j = t * 4; j < DN; j += 256 * 4) {
        float4 v = *(const float4*)(r + j);
        mn = fminf(mn, fminf(fminf(v.x, v.y), fminf(v.z, v.w)));
        mx = fmaxf(mx, fmaxf(fmaxf(v.x, v.y), fmaxf(v.z, v.w)));
        sm += v.x + v.y + v.z + v.w;
        sq += v.x * v.x + v.y * v.y + v.z * v.z + v.w * v.w;
    }
    __shared__ float smn[256], smx[256], ssm[256], ssq[256];
    smn[t] = mn; smx[t] = mx; ssm[t] = sm; ssq[t] = sq;
    __syncthreads();
    for (int s = 128; s > 0; s >>= 1) {
        if (t < s) {
            smn[t] = fminf(smn[t], smn[t + s]);
            smx[t] = fmaxf(smx[t], smx[t + s]);
            ssm[t] += ssm[t + s];
            ssq[t] += ssq[t + s];
        }
        __syncthreads();
    }
    if (t == 0) {
        float gmin = smn[0], gmax = smx[0], S1 = ssm[0], S2 = ssq[0];
        float scale = 1.0f / (gmax - gmin);
        // ||ngrad||^2 = scale^2 * (S2 - 2*gmin*S1 + D*gmin^2)
        float n2  = scale * scale * (S2 - 2.0f * gmin * S1 + (float)DN * gmin * gmin);
        float nrm = sqrtf(fmaxf(n2, 0.0f));
        float inv = 1.0f / fmaxf(nrm, 1e-8f);   // torch CosineSimilarity eps
        alpha[row] = scale * inv;
        beta[row]  = gmin * scale * inv;
    }
}

// ---------------------------------------------------------------------------
// K2: pred = argmax(outputs), per-block CE partial sums
// ---------------------------------------------------------------------------
__global__ void k2_pred_ce(const float* __restrict__ outp,
                           const int* __restrict__ y,
                           int* __restrict__ pred,
                           float* __restrict__ ce_part) {
    const int i = blockIdx.x * 256 + threadIdx.x;
    const float* o = outp + (size_t)i * NCLS;
    float m = o[0]; int am = 0;
    #pragma unroll
    for (int c = 1; c < NCLS; ++c) { float v = o[c]; if (v > m) { m = v; am = c; } }
    float s = 0.f;
    #pragma unroll
    for (int c = 0; c < NCLS; ++c) s += expf(o[c] - m);
    float ce = m + logf(s) - o[y[i]];           // -log softmax[y]
    pred[i] = am;

    __shared__ float red[256];
    red[threadIdx.x] = ce;
    __syncthreads();
    for (int s2 = 128; s2 > 0; s2 >>= 1) {
        if (threadIdx.x < s2) red[threadIdx.x] += red[threadIdx.x + s2];
        __syncthreads();
    }
    if (threadIdx.x == 0) ce_part[blockIdx.x] = red[0];
}

// ---------------------------------------------------------------------------
// K2.5: materialize the WMMA A-fragments once (they depend only on (lane,k)).
//   Afrag[kstep][lane] = { P(m, i0), P(m, i1) },  P(c,i) = (pred_i==c)?alpha_i:0
//   with m = lane&15, hl = lane>>4, i0 = 4*kstep + 2*hl, i1 = i0+1.
// 256 KB total -> L2-resident, shared by all 256 GEMM waves.
// ---------------------------------------------------------------------------
__global__ void k25_build_afrag(const float* __restrict__ alpha,
                                const int* __restrict__ pred,
                                float* __restrict__ afrag) {
    const int idx   = blockIdx.x * 256 + threadIdx.x;  // (BN/4)*32 entries
    const int lane  = idx & 31;
    const int kstep = idx >> 5;
    const int m  = lane & 15;
    const int hl = lane >> 4;
    const int i0 = kstep * 4 + 2 * hl;
    const int i1 = i0 + 1;
    // unconditional loads, arithmetic select (no predicated memory ops)
    float al0 = alpha[i0], al1 = alpha[i1];
    int   p0  = pred[i0],  p1  = pred[i1];
    float a0 = (p0 == m) ? al0 : 0.0f;
    float a1 = (p1 == m) ? al1 : 0.0f;
    float2 v; v.x = a0; v.y = a1;
    ((float2*)afrag)[idx] = v;
}

// ---------------------------------------------------------------------------
// K3: Sraw[16 x 4096] = P^T * G via V_WMMA_F32_16X16X4_F32
// 256 threads = 8 waves; each wave owns one 16-column tile of S.
// Inner loop: 1x global_load_b64 (A) + 2x global_load_b32 (B) + 1x WMMA.
// ---------------------------------------------------------------------------
__global__ void k3_class_gemm(const float* __restrict__ g,
                              const float* __restrict__ afrag,
                              float* __restrict__ S) {
    const int wave = threadIdx.x >> 5;
    const int lane = threadIdx.x & 31;
    const int tile = blockIdx.x * (blockDim.x >> 5) + wave;
    const int d0   = tile * 16;
    const int m    = lane & 15;   // B: d column
    const int hl   = lane >> 4;   // lane half selects K pair {0,1} vs {2,3}

    v8f acc = {};
    const float* __restrict__ gcol = g + d0 + m;
    const v2f*   __restrict__ ap   = (const v2f*)afrag + lane;  // +32 per kstep

    #pragma unroll 4
    for (int k = 0; k < BN; k += 4) {
        const int i0 = k + 2 * hl;
        v2f a = ap[(k >> 2) * 32];
        v2f b;
        b.x = gcol[(size_t)i0 * DN];
        b.y = gcol[(size_t)(i0 + 1) * DN];
        acc = __builtin_amdgcn_wmma_f32_16x16x4_f32(
            /*neg_a=*/false, a, /*neg_b=*/false, b,
            /*c_mod=*/(short)0, acc, /*reuse_a=*/false, /*reuse_b=*/false);
    }

    // C/D layout: VGPR r, lane -> row = r + 8*(lane>>4), col = lane&15
    float* __restrict__ Sd = S + d0 + m;
    #pragma unroll
    for (int r = 0; r < 8; ++r)
        Sd[(size_t)(r + 8 * hl) * DN] = acc[r];
}

// ---------------------------------------------------------------------------
// K4: finalize (single block, fully deterministic reductions)
//  S_c[d] = Sraw_c[d] - beta_c  =>  ||S_c||^2 = sq_c - 2*beta_c*sum_c + D*beta_c^2
//  loss = CE/B + sum_c (n_c^2 - ||S_c||^2) / (2B)
// ---------------------------------------------------------------------------
__global__ void k4_finalize(const float* __restrict__ S,
                            const float* __restrict__ beta,
                            const int* __restrict__ pred,
                            const float* __restrict__ ce_part, int nce,
                            float* __restrict__ out) {
    const int t = threadIdx.x;          // 256 threads: 16 classes x 16 slices
    const int c   = t >> 4;
    const int sub = t & 15;

    float lb = 0.f, lc = 0.f;
    for (int i = sub; i < BN; i += 16) {
        float bv = beta[i];
        int   pv = pred[i];
        if (pv == c) { lb += bv; lc += 1.0f; }
    }
    float ls = 0.f, lq = 0.f;
    const float* Sc = S + (size_t)c * DN;
    for (int d = sub; d < DN; d += 16) {
        float v = Sc[d];
        ls += v;
        lq += v * v;
    }
    __shared__ float pb[256], pc[256], ps[256], pq[256];
    pb[t] = lb; pc[t] = lc; ps[t] = ls; pq[t] = lq;
    __syncthreads();

    if (t == 0) {
        float ce = 0.f;
        for (int j = 0; j < nce; ++j) ce += ce_part[j];
        float x = 0.f;
        for (int cc = 0; cc < NCLS; ++cc) {
            float n = 0.f, bt = 0.f, sm = 0.f, sq = 0.f;
            for (int s = 0; s < 16; ++s) {
                int idx = cc * 16 + s;
                n += pc[idx]; bt += pb[idx]; sm += ps[idx]; sq += pq[idx];
            }
            float normsq = sq - 2.0f * bt * sm + (float)DN * bt * bt;
            x += n * n - normsq;
        }
        out[0] = ce / (float)BN + x / (2.0f * (float)BN);
    }
}

// ---------------------------------------------------------------------------
extern "C" void kernel_launch(void* const* d_in, const int* in_sizes, int n_in,
                              void* d_out, int out_size, void* d_ws, size_t ws_size,
                              hipStream_t stream) {
    const float* outputs = (const float*)d_in[0];   // [4096,10] f32
    const float* grad    = (const float*)d_in[1];   // [4096,1,64,64] f32
    const int*   y       = (const int*)d_in[2];     // [4096] i32

    float* ws      = (float*)d_ws;
    float* alpha   = ws;                    // 4096 floats
    float* beta    = ws + BN;               // 4096 floats
    int*   pred    = (int*)(ws + 2 * BN);   // 4096 ints
    float* ce_part = ws + 3 * BN;           // 16 floats (padded to 64)
    float* S       = ws + 3 * BN + 64;      // 16*4096 floats
    float* afrag   = S + 16 * DN;           // (BN/4)*32*2 floats = 256 KB

    k1_row_stats <<<BN, 256, 0, stream>>>(grad, alpha, beta);
    k2_pred_ce   <<<BN / 256, 256, 0, stream>>>(outputs, y, pred, ce_part);
    k25_build_afrag<<<(BN / 4) * 32 / 256, 256, 0, stream>>>(alpha, pred, afrag);
    k3_class_gemm<<<(DN / 16) / 8, 256, 0, stream>>>(grad, afrag, S);
    k4_finalize  <<<1, 256, 0, stream>>>(S, beta, pred, ce_part, BN / 256,
                                         (float*)d_out);
}